// HetGCNLayer_37254546325572
// MI455X (gfx1250) — compile-verified
//
#include <hip/hip_runtime.h>

#define H 8
#define D 16
#define HD 128
#define NEG_SLOPE 0.2f

typedef __attribute__((ext_vector_type(2))) float v2f;
typedef __attribute__((ext_vector_type(8))) float v8f;

// ---------------------------------------------------------------------------
// Order-preserving float<->uint mapping so segment-max can use native u32
// atomic max (GLOBAL_ATOMIC_MAX_U32) instead of a CAS loop.
// ---------------------------------------------------------------------------
__device__ __forceinline__ unsigned float_to_ordered(float f) {
    unsigned b = __float_as_uint(f);
    return (b & 0x80000000u) ? ~b : (b | 0x80000000u);
}
__device__ __forceinline__ float ordered_to_float(unsigned m) {
    unsigned b = (m & 0x80000000u) ? (m & 0x7FFFFFFFu) : ~m;
    return __uint_as_float(b);
}

// Force the native GLOBAL_ATOMIC_ADD_F32 (no CAS loop). unsafeAtomicAdd is
// HIP's documented hardware-FP-atomic intrinsic on AMD.
__device__ __forceinline__ void atomic_add_f32_native(float* p, float v) {
    unsafeAtomicAdd(p, v);
}

// ---------------------------------------------------------------------------
// Phase 1: per-node attention logits via V_WMMA_F32_16X16X4_F32.
//   s[n,h] = leaky_relu( sum_d hs[n, h*16+d] * attn[h*16+d] )
// Viewed as C[16x16] = A[16x128] * B[128x8pad16], B block-diagonal selector.
// One wave per 16-node tile; K=128 consumed as 32 chained K=4 WMMAs.
// A layout (32-bit 16x4): lane M = lane&15; float2 = {K=k0+h2, K=k0+h2+1},
//   h2 = (lane>=16)*2.  B layout (4x16) mirrors it with N = lane&15.
// C layout: VGPR r holds rows M=r (lanes 0-15) / M=r+8 (lanes 16-31), N=lane&15.
// ---------------------------------------------------------------------------
__global__ __launch_bounds__(128) void gat_node_scores(
    const float* __restrict__ hs, const float* __restrict__ attn,
    float* __restrict__ s, int N) {
    int lane = threadIdx.x & 31;
    int tile = blockIdx.x * 4 + (threadIdx.x >> 5);
    int n    = lane & 15;             // A row M / B-C col N
    int h2   = (lane >> 4) << 1;      // 0 or 2: K offset for this lane-half

    int row  = tile * 16 + n;
    int rowc = row < N ? row : N - 1; // clamp tail-tile loads (safe dup)
    const float* __restrict__ arow = hs + (size_t)rowc * HD;

    v8f c = {0.f, 0.f, 0.f, 0.f, 0.f, 0.f, 0.f, 0.f};
#pragma unroll
    for (int k0 = 0; k0 < HD; k0 += 4) {
        int k = k0 + h2;
        v2f a;
        a.x = arow[k];
        a.y = arow[k + 1];
        // B[k][n] = attn[k] iff n == k>>4 (head selector), else 0.
        float w0 = attn[k], w1 = attn[k + 1];
        int  tgt = k >> 4;            // same for k and k+1 (k0 % 4 == 0)
        v2f b;
        b.x = (n == tgt) ? w0 : 0.f;
        b.y = (n == tgt) ? w1 : 0.f;
        c = __builtin_amdgcn_wmma_f32_16x16x4_f32(
            /*neg_a=*/false, a, /*neg_b=*/false, b,
            /*c_mod=*/(short)0, c, /*reuse_a=*/false, /*reuse_b=*/false);
    }

    if (n < H) {                      // columns 8..15 are padding (zero B)
        int mbase = (lane >> 4) * 8;
#pragma unroll
        for (int r = 0; r < 8; ++r) {
            int node = tile * 16 + mbase + r;
            if (node < N) {
                float el = c[r];
                s[node * H + n] = el > 0.f ? el : NEG_SLOPE * el;
            }
        }
    }
}

// ---------------------------------------------------------------------------
// Phase 2: segment max over incoming edges (native u32 atomic max).
// One thread per (edge, head).
// ---------------------------------------------------------------------------
__global__ __launch_bounds__(256) void gat_edge_max(
    const float* __restrict__ s, const int* __restrict__ src,
    const int* __restrict__ dst, unsigned* __restrict__ segmax, int E) {
    int i = blockIdx.x * blockDim.x + threadIdx.x;
    if (i >= E * H) return;
    int e = i >> 3, h = i & 7;
    float v = s[src[e] * H + h];
    atomicMax(&segmax[dst[e] * H + h], float_to_ordered(v));
}

// ---------------------------------------------------------------------------
// Phase 3: denom[dst,h] = sum_e exp(s - max)   (native f32 atomic add).
// ---------------------------------------------------------------------------
__global__ __launch_bounds__(256) void gat_edge_expsum(
    const float* __restrict__ s, const int* __restrict__ src,
    const int* __restrict__ dst, const unsigned* __restrict__ segmax,
    float* __restrict__ denom, int E) {
    int i = blockIdx.x * blockDim.x + threadIdx.x;
    if (i >= E * H) return;
    int e = i >> 3, h = i & 7;
    int de = dst[e];
    float v  = s[src[e] * H + h];
    float mx = ordered_to_float(segmax[de * H + h]);
    atomic_add_f32_native(&denom[de * H + h], __expf(v - mx));
}

// ---------------------------------------------------------------------------
// Phase 4: out[dst] += hs[src] * a.  One thread per (edge, float4-chunk):
// 32 threads (one wave) per edge; b128 coalesced gather of hs[src];
// 4 x GLOBAL_ATOMIC_ADD_F32 scatter.  a recomputed with the same __expf as
// phase 3 so the softmax weights are self-consistent.
// ---------------------------------------------------------------------------
__global__ __launch_bounds__(256) void gat_edge_aggregate(
    const float* __restrict__ hs, const float* __restrict__ s,
    const int* __restrict__ src, const int* __restrict__ dst,
    const unsigned* __restrict__ segmax, const float* __restrict__ denom,
    float* __restrict__ out, int E) {
    int i = blockIdx.x * blockDim.x + threadIdx.x;
    if (i >= E * 32) return;
    int e = i >> 5;
    int q = i & 31;          // float4 chunk within the 128-wide feature row
    int h = q >> 2;          // head = (q*4)/16

    int se = src[e], de = dst[e];
    float v   = s[se * H + h];
    float mx  = ordered_to_float(segmax[de * H + h]);
    float den = denom[de * H + h];
    float a   = __expf(v - mx) / den;

    const float4 hv = *(const float4*)(hs + (size_t)se * HD + q * 4);
    float* o = out + (size_t)de * HD + q * 4;
    atomic_add_f32_native(o + 0, hv.x * a);
    atomic_add_f32_native(o + 1, hv.y * a);
    atomic_add_f32_native(o + 2, hv.z * a);
    atomic_add_f32_native(o + 3, hv.w * a);
}

// ---------------------------------------------------------------------------
extern "C" void kernel_launch(void* const* d_in, const int* in_sizes, int n_in,
                              void* d_out, int out_size, void* d_ws, size_t ws_size,
                              hipStream_t stream) {
    const float* h_src = (const float*)d_in[0];
    // d_in[1] (h_dst) is unused by the reference computation.
    const float* attn  = (const float*)d_in[2];   // 128 floats, [1,8,16] flat
    const int*   src   = (const int*)d_in[3];
    const int*   dst   = (const int*)d_in[4];

    int N_src = in_sizes[0] / HD;
    int N_dst = in_sizes[1] / HD;
    int E     = in_sizes[3];

    // Workspace layout: s[N_src*8] f32 | segmax[N_dst*8] u32 | denom[N_dst*8] f32
    float*    sbuf   = (float*)d_ws;
    unsigned* segmax = (unsigned*)(sbuf + (size_t)N_src * H);
    float*    denom  = (float*)(segmax + (size_t)N_dst * H);
    float*    out    = (float*)d_out;

    // Atomics accumulate: zero destinations every call (deterministic replay).
    // segmax init 0 == ordered encoding below every real float value.
    hipMemsetAsync(out,    0, (size_t)out_size * sizeof(float), stream);
    hipMemsetAsync(segmax, 0, (size_t)N_dst * H * sizeof(unsigned), stream);
    hipMemsetAsync(denom,  0, (size_t)N_dst * H * sizeof(float), stream);

    int tiles = (N_src + 15) / 16;
    gat_node_scores<<<(tiles + 3) / 4, 128, 0, stream>>>(h_src, attn, sbuf, N_src);

    int t1 = E * H;
    gat_edge_max<<<(t1 + 255) / 256, 256, 0, stream>>>(sbuf, src, dst, segmax, E);
    gat_edge_expsum<<<(t1 + 255) / 256, 256, 0, stream>>>(sbuf, src, dst, segmax, denom, E);

    long long t2 = (long long)E * 32;
    gat_edge_aggregate<<<(unsigned)((t2 + 255) / 256), 256, 0, stream>>>(
        h_src, sbuf, src, dst, segmax, denom, out, E);
}